// AFDecoder_72112500899949
// MI455X (gfx1250) — compile-verified
//
#include <hip/hip_runtime.h>
#include <hip/hip_bf16.h>
#include <math.h>

#define DIM   384
#define N2    (DIM * DIM)          // 147456
#define NIMG  64
#define NATOM 8192
#define KW    11                   // gaussian cutoff width
#define HALF_K 5
#define LSTR  388                  // padded LDS row stride (banks: 388%64=4)

typedef float v2f __attribute__((ext_vector_type(2)));
typedef float v8f __attribute__((ext_vector_type(8)));

// ---------------------------------------------------------------------------
// fp32 WMMA 16x16x4 wrapper (D = A(16x4) * B(4x16) + C)
// ---------------------------------------------------------------------------
__device__ __forceinline__ v8f wmma_f32(v2f a, v2f b, v8f c) {
    return __builtin_amdgcn_wmma_f32_16x16x4_f32(
        /*neg_a=*/false, a, /*neg_b=*/false, b,
        /*c_mod=*/(short)0, c, /*reuse_a=*/false, /*reuse_b=*/false);
}

// ---------------------------------------------------------------------------
// Precompute shifted-DFT matrices:
//   Mc[k,n] = (-1)^(k+n) cos(2*pi*k*n/384)
//   Ms[k,n] = (-1)^(k+n) sin(2*pi*k*n/384)
//   Mp = Mc + Ms,  Mm = Mc - Ms
// (both fftshifts fold exactly into the sign: 192^2/384 is an integer phase)
// ---------------------------------------------------------------------------
__global__ void make_mats_kernel(float* __restrict__ Mc, float* __restrict__ Ms,
                                 float* __restrict__ Mp, float* __restrict__ Mm) {
    int idx = blockIdx.x * blockDim.x + threadIdx.x;
    if (idx >= N2) return;
    int k = idx / DIM;
    int n = idx - k * DIM;
    int kn = (k * n) % DIM;                       // exact: phase periodic in kn/384
    float s = ((k + n) & 1) ? -1.0f : 1.0f;
    double ang = (double)kn * (6.283185307179586 / (double)DIM);
    float c  = s * (float)cos(ang);
    float sn = s * (float)sin(ang);
    Mc[idx] = c;
    Ms[idx] = sn;
    Mp[idx] = c + sn;
    Mm[idx] = c - sn;
}

__global__ void zero_kernel(float* __restrict__ p, int n) {
    int idx = blockIdx.x * blockDim.x + threadIdx.x;
    if (idx < n) p[idx] = 0.0f;
}

// ---------------------------------------------------------------------------
// Gaussian splat: one thread per (image, atom). Pose transform then 11x11
// stamp via global f32 atomics into y_real.
// ---------------------------------------------------------------------------
__global__ void splat_kernel(const float* __restrict__ crd,
                             const float* __restrict__ rot,
                             const float* __restrict__ rot_init,
                             const float* __restrict__ trans_init,
                             float* __restrict__ y_real) {
    int tid = blockIdx.x * blockDim.x + threadIdx.x;
    if (tid >= NIMG * NATOM) return;
    int b = tid >> 13;                       // tid / 8192

    const float* p = crd + (size_t)tid * 3;
    float p0 = p[0], p1 = p[1], p2 = p[2];

    // crd @ rot_init + trans_init   (rot_init row-major [m][j])
    float c0 = p0 * rot_init[0] + p1 * rot_init[3] + p2 * rot_init[6] + trans_init[0];
    float c1 = p0 * rot_init[1] + p1 * rot_init[4] + p2 * rot_init[7] + trans_init[1];
    float c2 = p0 * rot_init[2] + p1 * rot_init[5] + p2 * rot_init[8] + trans_init[2];

    // einsum('bnj,bkj->bnk'): coordinate k = dot(c, rot[b][k][:])
    const float* R = rot + (size_t)b * 9;
    float X = c0 * R[0] + c1 * R[1] + c2 * R[2];
    float Y = c0 * R[3] + c1 * R[4] + c2 * R[5];

    float cx = X + (float)(DIM / 2);          // PIX = 1.0
    float cy = Y + (float)(DIM / 2);
    int icx = (int)rintf(cx);                 // round-half-to-even = jnp.round
    int icy = (int)rintf(cy);

    const float inv2s2 = 1.0f / (2.0f * 1.5f * 1.5f);

    float gx[KW], gy[KW];
    int   ixc[KW], iyc[KW];
    bool  mx[KW], my[KW];
#pragma unroll
    for (int o = 0; o < KW; ++o) {
        int xx = icx + o - HALF_K;
        int yy = icy + o - HALF_K;
        float dx = ((float)xx - (float)(DIM / 2)) - X;
        float dy = ((float)yy - (float)(DIM / 2)) - Y;
        gx[o] = expf(-(dx * dx) * inv2s2);
        gy[o] = expf(-(dy * dy) * inv2s2);
        mx[o] = (xx >= 0) && (xx < DIM);
        my[o] = (yy >= 0) && (yy < DIM);
        ixc[o] = min(max(xx, 0), DIM - 1);
        iyc[o] = min(max(yy, 0), DIM - 1);
    }

    float* img = y_real + (size_t)b * N2;
#pragma unroll
    for (int j = 0; j < KW; ++j) {
        if (!my[j]) continue;                // reference adds exact 0 here
        float gyv = gy[j];
        int rowoff = iyc[j] * DIM;
#pragma unroll
        for (int i = 0; i < KW; ++i) {
            if (!mx[i]) continue;
            atomicAdd(&img[rowoff + ixc[i]], gyv * gx[i]);
        }
    }
}

// ---------------------------------------------------------------------------
// Stage 1: A = Mc * X,  B = Ms * X   (per image)
// Block = 4 waves (128 thr). Mc/Ms 16-row strips staged in padded LDS.
// Each wave: 2 column tiles x 2 outputs = 4 accumulators; A-frag reuse x2,
// X-frag reuse x2.  Grid = (3, 24, imgs).
// ---------------------------------------------------------------------------
__global__ __launch_bounds__(128) void stage1_kernel(
    const float* __restrict__ Xall, const float* __restrict__ Mc,
    const float* __restrict__ Ms, float* __restrict__ A, float* __restrict__ B) {
    __shared__ float sMc[16 * LSTR];
    __shared__ float sMs[16 * LSTR];

    int tid  = threadIdx.x;
    int wave = tid >> 5;
    int lane = tid & 31;
    int l15  = lane & 15;
    int half = lane >> 4;
    int it   = blockIdx.y;
    int img  = blockIdx.z;

    // cooperative strip copy (rows contiguous in global; padded in LDS)
    {
        const float4* gc = (const float4*)(Mc + (size_t)(it * 16) * DIM);
        const float4* gs = (const float4*)(Ms + (size_t)(it * 16) * DIM);
        for (int i = tid; i < 16 * (DIM / 4); i += 128) {
            int row = i / (DIM / 4);
            int c4  = i - row * (DIM / 4);
            ((float4*)(sMc + row * LSTR))[c4] = gc[i];
            ((float4*)(sMs + row * LSTR))[c4] = gs[i];
        }
    }
    __syncthreads();

    const float* X = Xall + (size_t)img * N2;
    int jt0  = blockIdx.x * 8 + wave * 2;     // first of 2 column tiles
    int col0 = jt0 * 16 + l15;
    int col1 = col0 + 16;

    v8f accA0 = {0.f,0.f,0.f,0.f,0.f,0.f,0.f,0.f};
    v8f accA1 = accA0, accB0 = accA0, accB1 = accA0;

    for (int k0 = 0; k0 < DIM; k0 += 4) {
        int kc = k0 + (half << 1);
        v2f a1 = *(const v2f*)(sMc + l15 * LSTR + kc);   // ds_load_b64
        v2f a2 = *(const v2f*)(sMs + l15 * LSTR + kc);
        v2f b0, b1;
        b0.x = X[(size_t)kc * DIM + col0];
        b0.y = X[(size_t)(kc + 1) * DIM + col0];
        b1.x = X[(size_t)kc * DIM + col1];
        b1.y = X[(size_t)(kc + 1) * DIM + col1];
        accA0 = wmma_f32(a1, b0, accA0);
        accB0 = wmma_f32(a2, b0, accB0);
        accA1 = wmma_f32(a1, b1, accA1);
        accB1 = wmma_f32(a2, b1, accB1);
    }

    float* Ao = A + (size_t)img * N2;
    float* Bo = B + (size_t)img * N2;
    int r0 = it * 16 + half * 8;              // C/D layout: lanes 16-31 = M+8
#pragma unroll
    for (int r = 0; r < 8; ++r) {
        size_t off = (size_t)(r0 + r) * DIM;
        Ao[off + col0] = accA0[r];
        Ao[off + col1] = accA1[r];
        Bo[off + col0] = accB0[r];
        Bo[off + col1] = accB1[r];
    }
}

// ---------------------------------------------------------------------------
// Stage 2: y = A * Mp + B * Mm  (Hartley output, shifts pre-baked)
// Block = 3 waves (96 thr). A/B 16-row strips staged in padded LDS.
// Each wave: 4 column tiles, one accumulator each; A/B-frag reuse x4.
// Grid = (2, 24, imgs).
// ---------------------------------------------------------------------------
__global__ __launch_bounds__(96) void stage2_kernel(
    const float* __restrict__ A, const float* __restrict__ B,
    const float* __restrict__ Mp, const float* __restrict__ Mm,
    float* __restrict__ Yall) {
    __shared__ float sA[16 * LSTR];
    __shared__ float sB[16 * LSTR];

    int tid  = threadIdx.x;
    int wave = tid >> 5;                      // 0..2
    int lane = tid & 31;
    int l15  = lane & 15;
    int half = lane >> 4;
    int it   = blockIdx.y;
    int img  = blockIdx.z;

    {
        const float4* ga = (const float4*)(A + (size_t)img * N2 + (size_t)(it * 16) * DIM);
        const float4* gb = (const float4*)(B + (size_t)img * N2 + (size_t)(it * 16) * DIM);
        for (int i = tid; i < 16 * (DIM / 4); i += 96) {
            int row = i / (DIM / 4);
            int c4  = i - row * (DIM / 4);
            ((float4*)(sA + row * LSTR))[c4] = ga[i];
            ((float4*)(sB + row * LSTR))[c4] = gb[i];
        }
    }
    __syncthreads();

    int jt0 = blockIdx.x * 12 + wave * 4;     // first of 4 column tiles

    v8f acc0 = {0.f,0.f,0.f,0.f,0.f,0.f,0.f,0.f};
    v8f acc1 = acc0, acc2 = acc0, acc3 = acc0;

    for (int k0 = 0; k0 < DIM; k0 += 4) {
        int kc = k0 + (half << 1);
        v2f a1 = *(const v2f*)(sA + l15 * LSTR + kc);
        v2f a2 = *(const v2f*)(sB + l15 * LSTR + kc);
#pragma unroll
        for (int c = 0; c < 4; ++c) {
            int col = (jt0 + c) * 16 + l15;
            v2f bp, bm;
            bp.x = Mp[(size_t)kc * DIM + col];
            bp.y = Mp[(size_t)(kc + 1) * DIM + col];
            bm.x = Mm[(size_t)kc * DIM + col];
            bm.y = Mm[(size_t)(kc + 1) * DIM + col];
            v8f t = (c == 0) ? acc0 : (c == 1) ? acc1 : (c == 2) ? acc2 : acc3;
            t = wmma_f32(a1, bp, t);
            t = wmma_f32(a2, bm, t);
            if (c == 0) acc0 = t; else if (c == 1) acc1 = t;
            else if (c == 2) acc2 = t; else acc3 = t;
        }
    }

    float* Yo = Yall + (size_t)img * N2;
    int r0 = it * 16 + half * 8;
#pragma unroll
    for (int r = 0; r < 8; ++r) {
        size_t off = (size_t)(r0 + r) * DIM;
        Yo[off + (jt0 + 0) * 16 + l15] = acc0[r];
        Yo[off + (jt0 + 1) * 16 + l15] = acc1[r];
        Yo[off + (jt0 + 2) * 16 + l15] = acc2[r];
        Yo[off + (jt0 + 3) * 16 + l15] = acc3[r];
    }
}

// ---------------------------------------------------------------------------
// Host-side launch
// ---------------------------------------------------------------------------
extern "C" void kernel_launch(void* const* d_in, const int* in_sizes, int n_in,
                              void* d_out, int out_size, void* d_ws, size_t ws_size,
                              hipStream_t stream) {
    const float* crd        = (const float*)d_in[0];  // [64, 8192, 3]
    const float* rot        = (const float*)d_in[1];  // [64, 3, 3]
    const float* rot_init   = (const float*)d_in[2];  // [3, 3]
    const float* trans_init = (const float*)d_in[3];  // [3]

    float* out    = (float*)d_out;
    float* y      = out;                       // [64, 384, 384] hartley
    float* y_real = out + (size_t)NIMG * N2;   // [64, 384, 384] real image

    // Workspace layout: [Mc | Ms | Mp | Mm | A(chunk) | B(chunk)]
    float* ws = (float*)d_ws;
    float* Mc = ws;
    float* Ms = ws + (size_t)N2;
    float* Mp = ws + (size_t)2 * N2;
    float* Mm = ws + (size_t)3 * N2;
    float* AB = ws + (size_t)4 * N2;

    size_t matBytes = (size_t)4 * N2 * sizeof(float);
    size_t perImg   = (size_t)2 * N2 * sizeof(float);   // A + B per image
    int chunk = 1;
    if (ws_size > matBytes) {
        size_t c = (ws_size - matBytes) / perImg;
        chunk = (c >= NIMG) ? NIMG : (c < 1 ? 1 : (int)c);
    }

    // 1) DFT matrices (shifts baked in)
    make_mats_kernel<<<(N2 + 255) / 256, 256, 0, stream>>>(Mc, Ms, Mp, Mm);

    // 2) zero real-space images
    {
        int n = NIMG * N2;
        zero_kernel<<<(n + 255) / 256, 256, 0, stream>>>(y_real, n);
    }

    // 3) pose + gaussian splat (atomic scatter)
    {
        int n = NIMG * NATOM;
        splat_kernel<<<(n + 255) / 256, 256, 0, stream>>>(crd, rot, rot_init,
                                                          trans_init, y_real);
    }

    // 4) Hartley transform as two WMMA GEMM stages, chunked over images
    for (int b0 = 0; b0 < NIMG; b0 += chunk) {
        int imgs = NIMG - b0;
        if (imgs > chunk) imgs = chunk;
        float* A = AB;
        float* B = AB + (size_t)imgs * N2;
        dim3 g1(3, 24, imgs);
        stage1_kernel<<<g1, 128, 0, stream>>>(y_real + (size_t)b0 * N2, Mc, Ms, A, B);
        dim3 g2(2, 24, imgs);
        stage2_kernel<<<g2, 96, 0, stream>>>(A, B, Mp, Mm, y + (size_t)b0 * N2);
    }
}